// KPConvLayer_29489245454560
// MI455X (gfx1250) — compile-verified
//
#include <hip/hip_runtime.h>

// ---------------------------------------------------------------------------
// KPConv for MI455X (gfx1250, wave32):
//   Y[b,r,e] = sum_{m,d} WF[b,r,m,d] * W[m,d,e]
//   WF[b,r,m,d] = sum_n max(0, 1-||rel-Q[m]||) * F[b, N[b,r,n], d]
// fp32 end-to-end. GEMM via V_WMMA_F32_16X16X4_F32. W staged into LDS by the
// Tensor Data Mover (tensor_load_to_lds, TENSORcnt double-buffered pipeline).
// ---------------------------------------------------------------------------

typedef __attribute__((ext_vector_type(2))) float        v2f;
typedef __attribute__((ext_vector_type(8))) float        v8f;
typedef __attribute__((ext_vector_type(4))) unsigned int v4u;
typedef __attribute__((ext_vector_type(8))) int          v8i;
typedef __attribute__((ext_vector_type(4))) int          v4i;

#define KB      4
#define RP      4096
#define KAPPA   16
#define DIN     64
#define DOUT    320
#define MQ      19

#define TPTS     32      // points per workgroup = 2 WMMA M-tiles
#define NMT      2       // M-tiles
#define NTHREADS 160     // 5 wave32
#define NWAVES   5
#define NT_PER_WAVE 4    // 20 Dout-tiles / 5 waves -> 64 cols per wave

// ---- LDS layout (float units) ----
#define H_STRIDE_N  20                      // 19 h + pad
#define H_STRIDE_P  (KAPPA * H_STRIDE_N)    // 320
#define H_FLOATS    (TPTS * H_STRIDE_P)     // 10240
#define WF_STRIDE_P 68                      // 64 d + 4 pad -> conflict-free A b64
#define WF_STRIDE_M (TPTS * WF_STRIDE_P)    // 2176
#define WF_FLOATS   (MQ * WF_STRIDE_M)      // 41344
#define IDX_INTS    (TPTS * KAPPA)          // 512
// W staging: per (buffer, wave) chunk = 32 rows x 64 cols, TDM-padded to
// row stride 72 dwords (pad 8 dwords after every 64) -> bank-conflict-free.
#define STG_ROWSTRIDE 72
#define STG_CHUNK_FLOATS (32 * STG_ROWSTRIDE)            // 2304
#define STG_FLOATS  (2 * NWAVES * STG_CHUNK_FLOATS)      // 23040
#define SMEM_FLOATS (H_FLOATS + WF_FLOATS + IDX_INTS + STG_FLOATS)
#define SMEM_BYTES  (SMEM_FLOATS * 4)                    // 300544

#define NCHUNKS (2 * MQ)   // 38: (m, 32-row half) chunks

// ---- TDM: load one 32x64 f32 tile of W into LDS with 8-dword row padding ----
__device__ __forceinline__ void tdm_load_w_chunk(const float* __restrict__ W,
                                                 int chunk, unsigned lds_byte,
                                                 int col0) {
    const int m     = chunk >> 1;
    const int rbase = (chunk & 1) * 32;
    const unsigned long long ga = (unsigned long long)(size_t)
        (W + (size_t)m * DIN * DOUT + (size_t)rbase * DOUT + col0);

    v4u g0;
    g0[0] = 1u;                                   // count=1, user descriptor
    g0[1] = lds_byte;                             // lds_addr (bytes)
    g0[2] = (unsigned)(ga & 0xFFFFFFFFu);         // global_addr[31:0]
    g0[3] = (unsigned)((ga >> 32) & 0x01FFFFFFu)  // global_addr[56:32]
          | (2u << 30);                           // type=2 (image)

    v8i g1;
    g1[0] = (2 << 16)        // data_size = 4B
          | (1 << 20)        // pad_enable
          | (5 << 22)        // pad_interval: 64 dwords
          | (7 << 25);       // pad_amount:   8 dwords
    g1[1] = (DOUT & 0xFFFF) << 16;                   // tensor_dim0[15:0]
    g1[2] = (DOUT >> 16) | (64 << 16);               // dim0[31:16] | tensor_dim1[15:0]
    g1[3] = (64 << 16);                              // dim1[31:16]=0 | tile_dim0 = 64
    g1[4] = 32;                                      // tile_dim1 = 32, tile_dim2 = 0
    g1[5] = DOUT;                                    // tensor_dim0_stride = 320
    g1[6] = 0;
    g1[7] = 0;

    const v4i z4 = (v4i){0, 0, 0, 0};
    const v8i z8 = (v8i){0, 0, 0, 0, 0, 0, 0, 0};
    // clang-23 / therock form: (g0, g1, g2, g3, extra, cpol)
    __builtin_amdgcn_tensor_load_to_lds(g0, g1, z4, z4, z8, 0);
}

__global__ __launch_bounds__(NTHREADS) void kpconv_f32_wmma_tdm(
    const float* __restrict__ X,    // [4][4096][3]
    const float* __restrict__ F,    // [4][4096][64]
    const int*   __restrict__ Nbr,  // [4][4096][16]
    const float* __restrict__ Q,    // [19][3]
    const float* __restrict__ W,    // [19][64][320]
    float*       __restrict__ Y)    // [4][4096][320]
{
    extern __shared__ float smem[];
    float* sH    = smem;                                   // [32][16][20]
    float* sWF   = smem + H_FLOATS;                        // [19][32][68]
    int*   sIdx  = (int*)(smem + H_FLOATS + WF_FLOATS);    // [32][16]
    float* sStg  = smem + H_FLOATS + WF_FLOATS + IDX_INTS; // [2][5][32][72]

    const int tid  = threadIdx.x;
    const int lane = tid & 31;
    const int wave = tid >> 5;

    const int tilesPerBatch = RP / TPTS;                   // 128
    const int b    = blockIdx.x / tilesPerBatch;
    const int r0   = (blockIdx.x % tilesPerBatch) * TPTS;
    const int col0 = wave * (NT_PER_WAVE * 16);            // this wave's Dout base

    // Kick off the first two TDM chunks now; they overlap phases 0-1.
    {
        const unsigned stg0 = (unsigned)(size_t)(sStg + (size_t)(0 * NWAVES + wave) * STG_CHUNK_FLOATS);
        const unsigned stg1 = (unsigned)(size_t)(sStg + (size_t)(1 * NWAVES + wave) * STG_CHUNK_FLOATS);
        tdm_load_w_chunk(W, 0, stg0, col0);
        tdm_load_w_chunk(W, 1, stg1, col0);
    }

    // ---------------- Phase 0: h weights + neighbor index cache -------------
    for (int q = tid; q < TPTS * KAPPA; q += NTHREADS) {
        const int p = q >> 4;
        const int n = q & 15;
        const int idx = Nbr[(b * RP + r0 + p) * KAPPA + n];
        sIdx[q] = idx;
        const float* xp = X + (size_t)(b * RP + r0 + p) * 3;
        const float* xn = X + (size_t)(b * RP + idx) * 3;
        const float rx = xn[0] - xp[0];
        const float ry = xn[1] - xp[1];
        const float rz = xn[2] - xp[2];
        float* hrow = sH + p * H_STRIDE_P + n * H_STRIDE_N;
        #pragma unroll
        for (int m = 0; m < MQ; ++m) {
            const float dx = rx - Q[m * 3 + 0];
            const float dy = ry - Q[m * 3 + 1];
            const float dz = rz - Q[m * 3 + 2];
            const float d  = sqrtf(fmaf(dx, dx, fmaf(dy, dy, dz * dz)));
            hrow[m] = fmaxf(0.0f, 1.0f - d);   // SIGMA = 1
        }
    }
    __syncthreads();

    // ---------------- Phase 1: WF[m][p][d] into LDS -------------------------
    // 1024 tasks = 32 points x 32 d-pairs. NTHREADS is a multiple of 32, so a
    // whole wave shares one p per iteration: sH reads broadcast, F coalesced.
    for (int task = tid; task < TPTS * 32; task += NTHREADS) {
        const int p  = task >> 5;
        const int d0 = (task & 31) * 2;
        float accx[MQ], accy[MQ];
        #pragma unroll
        for (int m = 0; m < MQ; ++m) { accx[m] = 0.0f; accy[m] = 0.0f; }
        const float* hbase = sH + p * H_STRIDE_P;
        const int*   ip    = sIdx + p * KAPPA;
        for (int n = 0; n < KAPPA; ++n) {
            const int idx = ip[n];
            const float2 f = *(const float2*)(F + (size_t)(b * RP + idx) * DIN + d0);
            const float* hrow = hbase + n * H_STRIDE_N;
            #pragma unroll
            for (int m = 0; m < MQ; ++m) {
                const float h = hrow[m];
                accx[m] = fmaf(h, f.x, accx[m]);
                accy[m] = fmaf(h, f.y, accy[m]);
            }
        }
        #pragma unroll
        for (int m = 0; m < MQ; ++m) {
            float2* wf = (float2*)(sWF + m * WF_STRIDE_M + p * WF_STRIDE_P + d0);
            *wf = make_float2(accx[m], accy[m]);
        }
    }
    __syncthreads();

    // ---------------- Phase 2: GEMM via V_WMMA_F32_16X16X4_F32 ---------------
    // A (16x4 f32): lane pl holds row M=pl; VGPR0 = K{0,2}, VGPR1 = K{1,3}
    //   -> one b64 LDS load per fragment from stride-68 WF (conflict-free).
    // B (4x16 f32): lane pl = column; rows r/r+1 at LDS stride 72
    //   -> ds_load_2addr_b32 (off1=72), conflict-free by construction.
    const int pl    = lane & 15;
    const int khalf = (lane >> 4) * 2;       // 0 or 2

    v8f acc[NMT][NT_PER_WAVE];
    #pragma unroll
    for (int mt = 0; mt < NMT; ++mt)
        #pragma unroll
        for (int j = 0; j < NT_PER_WAVE; ++j) acc[mt][j] = (v8f){};

    for (int c = 0; c < NCHUNKS; ++c) {
        const int m     = c >> 1;
        const int rbase = (c & 1) * 32;
        const int buf   = c & 1;

        // Wait for chunk c (in-order completion; chunk c+1 may stay in flight).
        if (c + 1 < NCHUNKS) __builtin_amdgcn_s_wait_tensorcnt(1);
        else                 __builtin_amdgcn_s_wait_tensorcnt(0);

        const float* stg = sStg + (size_t)(buf * NWAVES + wave) * STG_CHUNK_FLOATS;
        const float* wfm = sWF + m * WF_STRIDE_M + khalf;

        #pragma unroll
        for (int kk = 0; kk < 8; ++kk) {
            const int dbase = rbase + kk * 4;
            const v2f a0 = *(const v2f*)(wfm + (0 * 16 + pl) * WF_STRIDE_P + dbase);
            const v2f a1 = *(const v2f*)(wfm + (1 * 16 + pl) * WF_STRIDE_P + dbase);
            #pragma unroll
            for (int jj = 0; jj < NT_PER_WAVE; ++jj) {
                const float* sp = stg + (kk * 4 + khalf) * STG_ROWSTRIDE + jj * 16 + pl;
                v2f bf;
                bf.x = sp[0];
                bf.y = sp[STG_ROWSTRIDE];
                acc[0][jj] = __builtin_amdgcn_wmma_f32_16x16x4_f32(
                    false, a0, false, bf, (short)0, acc[0][jj], false, false);
                acc[1][jj] = __builtin_amdgcn_wmma_f32_16x16x4_f32(
                    false, a1, false, bf, (short)0, acc[1][jj], false, false);
            }
        }

        // Ensure our LDS reads of this buffer retired, then refill it.
        asm volatile("s_wait_dscnt 0x0" ::: "memory");
        if (c + 2 < NCHUNKS) {
            const unsigned stgn = (unsigned)(size_t)
                (sStg + (size_t)(buf * NWAVES + wave) * STG_CHUNK_FLOATS);
            tdm_load_w_chunk(W, c + 2, stgn, col0);
        }
    }

    // ---------------- Epilogue: C/D layout -> global Y -----------------------
    // VGPR v: lanes 0-15 -> (M=v, N=lane); lanes 16-31 -> (M=v+8, N=lane-16)
    const int mrow = (lane >> 4) * 8;
    #pragma unroll
    for (int mt = 0; mt < NMT; ++mt) {
        #pragma unroll
        for (int jj = 0; jj < NT_PER_WAVE; ++jj) {
            const int e0 = col0 + jj * 16;
            #pragma unroll
            for (int v = 0; v < 8; ++v) {
                Y[(size_t)(b * RP + r0 + mt * 16 + mrow + v) * DOUT + e0 + pl]
                    = acc[mt][jj][v];
            }
        }
    }
}

extern "C" void kernel_launch(void* const* d_in, const int* in_sizes, int n_in,
                              void* d_out, int out_size, void* d_ws, size_t ws_size,
                              hipStream_t stream) {
    (void)in_sizes; (void)n_in; (void)out_size; (void)d_ws; (void)ws_size;
    const float* X   = (const float*)d_in[0];
    const float* F   = (const float*)d_in[1];
    const int*   Nbr = (const int*)  d_in[2];
    const float* Q   = (const float*)d_in[3];
    const float* W   = (const float*)d_in[4];
    float*       Y   = (float*)d_out;

    const int blocks = KB * (RP / TPTS);   // 512 workgroups, 160 threads each
    kpconv_f32_wmma_tdm<<<blocks, NTHREADS, SMEM_BYTES, stream>>>(X, F, Nbr, Q, W, Y);
}